// CGMPBlock_77962246357671
// MI455X (gfx1250) — compile-verified
//
#include <hip/hip_runtime.h>

#define LMAXV 2
#define CCH 8

typedef float v2f __attribute__((ext_vector_type(2)));
typedef float v8f __attribute__((ext_vector_type(8)));

// ---------------- compile-time Clebsch-Gordan tables (Racah formula) --------
constexpr double cfact(int n) { double r = 1.0; for (int i = 2; i <= n; ++i) r *= i; return r; }
constexpr double csqrt_(double x) {
  if (x <= 0.0) return 0.0;
  double r = x < 1.0 ? 1.0 : x;
  for (int i = 0; i < 100; ++i) r = 0.5 * (r + x / r);
  return r;
}
constexpr double cgcoef(int j1, int m1, int j2, int m2, int j, int m) {
  if (m1 + m2 != m) return 0.0;
  int lo = j1 > j2 ? j1 - j2 : j2 - j1;
  if (j < lo || j > j1 + j2) return 0.0;
  double pre = csqrt_((2.0 * j + 1.0) * cfact(j + j1 - j2) * cfact(j - j1 + j2) *
                      cfact(j1 + j2 - j) / cfact(j1 + j2 + j + 1));
  pre *= csqrt_(cfact(j + m) * cfact(j - m) * cfact(j1 - m1) * cfact(j1 + m1) *
                cfact(j2 - m2) * cfact(j2 + m2));
  double s = 0.0;
  for (int k = 0; k <= j1 + j2 - j; ++k) {
    int d2 = j1 + j2 - j - k, d3 = j1 - m1 - k, d4 = j2 + m2 - k;
    int d5 = j - j2 + m1 + k, d6 = j - j1 - m2 + k;
    if (d2 < 0 || d3 < 0 || d4 < 0 || d5 < 0 || d6 < 0) continue;
    double den = cfact(k) * cfact(d2) * cfact(d3) * cfact(d4) * cfact(d5) * cfact(d6);
    s += ((k & 1) ? -1.0 : 1.0) / den;
  }
  return pre * s;
}

struct CGPack {
  int ncoup[3];
  int l1t[3][6], l2t[3][6];
  int nnz[3][5][6];       // [l][m][coupling]
  int pq[3][5][6][5];     // p*8+q
  float coef[3][5][6][5];
};

constexpr CGPack buildCG() {
  CGPack P{};
  for (int l = 0; l <= LMAXV; ++l) {
    int nc = 0;
    for (int l1 = 0; l1 <= LMAXV; ++l1)
      for (int l2 = 0; l2 <= LMAXV; ++l2) {
        int lo = l1 > l2 ? l1 - l2 : l2 - l1;
        int hi = (l1 + l2 > LMAXV) ? LMAXV : (l1 + l2);
        if (l < lo || l > hi) continue;
        P.l1t[l][nc] = l1; P.l2t[l][nc] = l2;
        for (int mi = 0; mi < 2 * l + 1; ++mi) {
          int m = mi - l, n = 0;
          for (int p = 0; p < 2 * l1 + 1; ++p) {
            int m1 = p - l1, m2 = m - m1;
            if (m2 < -l2 || m2 > l2) continue;
            double v = cgcoef(l1, m1, l2, m2, l, m);
            if (v == 0.0) continue;
            P.pq[l][mi][nc][n] = p * 8 + (m2 + l2);
            P.coef[l][mi][nc][n] = (float)v;
            ++n;
          }
          P.nnz[l][mi][nc] = n;
        }
        ++nc;
      }
    P.ncoup[l] = nc;
  }
  return P;
}

// constexpr (NOT __constant__): emitted as a non-externally-initialized LLVM
// constant, so after the l/prod/cp/m loops unroll, every table access folds to
// an immediate and the nnz trip counts become static.
constexpr CGPack H = buildCG();
constexpr int LOFF[3] = {0, 8, 32};
constexpr int NCOUPC[3] = {3, 6, 6};

// ---------------- kernel 1: zero the y accumulator --------------------------
__global__ void zero_f32(float* __restrict__ p, long n) {
  long i = (long)blockIdx.x * blockDim.x + threadIdx.x;
  if (i < n) p[i] = 0.0f;
}

// ---------------- kernel 2: edge gather-scale-scatter (message passing) -----
// y layout per node: [0..8)=l0, [8..32)=l1 (c*3+m), [32..72)=l2 (c*5+m)
// identical ordering to concat(x0,x1,x2) so src/dst offsets coincide.
// 6 threads per edge, 3 float4 slots each.
__global__ __launch_bounds__(256) void edge_mp(
    const float* __restrict__ x0, const float* __restrict__ x1,
    const float* __restrict__ x2, const float* __restrict__ ev,
    const int* __restrict__ src, const int* __restrict__ dst,
    float* __restrict__ y, int nE) {
  long idx = (long)blockIdx.x * blockDim.x + threadIdx.x;
  if (idx >= (long)nE * 6) return;
  int e = (int)(idx / 6);
  int g = (int)(idx % 6);
  int s = src[e], d = dst[e];
  float v = ev[e];
#pragma unroll
  for (int j = 0; j < 3; ++j) {
    int slot = g * 3 + j;
    const float* ps;
    if (slot < 2)      ps = x0 + (size_t)s * 8  + slot * 4;
    else if (slot < 8) ps = x1 + (size_t)s * 24 + (slot - 2) * 4;
    else               ps = x2 + (size_t)s * 40 + (slot - 8) * 4;
    float4 val = *(const float4*)ps;
    float* py = y + (size_t)d * 72 + slot * 4;
    __hip_atomic_fetch_add(py + 0, v * val.x, __ATOMIC_RELAXED, __HIP_MEMORY_SCOPE_AGENT);
    __hip_atomic_fetch_add(py + 1, v * val.y, __ATOMIC_RELAXED, __HIP_MEMORY_SCOPE_AGENT);
    __hip_atomic_fetch_add(py + 2, v * val.z, __ATOMIC_RELAXED, __HIP_MEMORY_SCOPE_AGENT);
    __hip_atomic_fetch_add(py + 3, v * val.w, __ATOMIC_RELAXED, __HIP_MEMORY_SCOPE_AGENT);
  }
}

// ---------------- kernel 3: fused CG products + SO3 linear via f32 WMMA -----
// 64 threads = 2 waves; each wave owns 16 nodes. Per l: GEMM rows=16 nodes,
// cols=8 outputs (cols 8..15 accumulate harmless garbage, never stored),
// K=ncp, accumulated with WMMA_F32_16X16X4_F32 per m. A elements are built
// on the fly from the sparse CG contraction with literal coefficients.
__global__ __launch_bounds__(64) void node_cg_wmma(
    const float* __restrict__ x0, const float* __restrict__ x1,
    const float* __restrict__ x2, const float* __restrict__ y,
    const float* __restrict__ wyy0, const float* __restrict__ wyy1,
    const float* __restrict__ wyy2, const float* __restrict__ wyx0,
    const float* __restrict__ wyx1, const float* __restrict__ wyx2,
    const float* __restrict__ wxx0, const float* __restrict__ wxx1,
    const float* __restrict__ wxx2, float* __restrict__ out, int nN) {
  __shared__ float lds[2 * 16 * 144];  // per wave: 16 nodes x (72 x | 72 y)

  const int wave = threadIdx.x >> 5;
  const int lane = threadIdx.x & 31;
  const int row  = lane & 15;   // node-row for A / output channel for B
  const int hi   = lane >> 4;   // K-half selector
  const int wcol = row & 7;     // clamped weight column (cols 8..15 = garbage)
  const int tileBase = blockIdx.x * 32 + wave * 16;

  const float* W[3][3] = {{wyy0, wyx0, wxx0}, {wyy1, wyx1, wxx1}, {wyy2, wyx2, wxx2}};
  __builtin_prefetch(W[0][0], 0, 3);
  __builtin_prefetch(W[1][0], 0, 3);
  __builtin_prefetch(W[2][0], 0, 3);

  // ---- stage x and y for this wave's 16 nodes into LDS ----
  float* L = lds + wave * 16 * 144;
  for (int i = lane; i < 16 * 72; i += 32) {
    int ln = i / 72, f = i % 72;
    int node = tileBase + ln;
    if (node >= nN) node = nN - 1;
    float xv;
    if (f < 8)       xv = x0[(size_t)node * 8  + f];
    else if (f < 32) xv = x1[(size_t)node * 24 + (f - 8)];
    else             xv = x2[(size_t)node * 40 + (f - 32)];
    L[ln * 144 + f]      = xv;
    L[ln * 144 + 72 + f] = y[(size_t)node * 72 + f];
  }
  __syncthreads();

  const float* Lrow = L + row * 144;
  const v8f vzero = {0.f, 0.f, 0.f, 0.f, 0.f, 0.f, 0.f, 0.f};

#pragma unroll
  for (int l = 0; l < 3; ++l) {
    const int nm = 2 * l + 1;
    v8f acc[5];
#pragma unroll
    for (int m = 0; m < 5; ++m) acc[m] = vzero;

#pragma unroll
    for (int prod = 0; prod < 3; ++prod) {
      const float* wl = W[l][prod];
      const int aoff = (prod < 2)  ? 72 : 0;  // a side: y,y,x
      const int boff = (prod == 0) ? 72 : 0;  // b side: y,x,x
#pragma unroll
      for (int cp = 0; cp < NCOUPC[l]; ++cp) {
        const int l1 = H.l1t[l][cp], l2 = H.l2t[l][cp];   // fold to constants
        const int na = 2 * l1 + 1, nb = 2 * l2 + 1;
        const float* paBase = Lrow + aoff + LOFF[l1];
        const float* pbBase = Lrow + boff + LOFF[l2];
        const float* wcp = wl + (size_t)(cp * 64) * CCH;
        for (int c = 0; c < 8; ++c) {
          const float* pa = paBase + c * na;
          for (int dc = 0; dc < 8; dc += 4) {
            const int dlane = dc + 2 * hi;                // this lane's d pair
            const int kb = c * 8 + dlane;                 // K row in coupling
            v2f B;
            B.x = wcp[(kb + 0) * CCH + wcol];             // unconditional loads
            B.y = wcp[(kb + 1) * CCH + wcol];
            const float* pb0 = pbBase + dlane * nb;
            const float* pb1 = pb0 + nb;
#pragma unroll
            for (int m = 0; m < 2 * l + 1; ++m) {
              float z0 = 0.f, z1 = 0.f;
#pragma unroll
              for (int t = 0; t < H.nnz[l][m][cp]; ++t) { // static trip count
                const int pq = H.pq[l][m][cp][t];         // folds to literal
                const float cf = H.coef[l][m][cp][t];     // folds to literal
                const float a = pa[pq >> 3];
                z0 += cf * a * pb0[pq & 7];
                z1 += cf * a * pb1[pq & 7];
              }
              v2f A; A.x = z0; A.y = z1;
              acc[m] = __builtin_amdgcn_wmma_f32_16x16x4_f32(
                  false, A, false, B, (short)0, acc[m], false, false);
            }
          }
        }
      }
    }

    // ---- add identity x, scatter D tile to output ----
    // D layout: VGPR r holds row (r + 8*hi), col = lane&15
    const int o = row;
    if (o < CCH) {
      const size_t baseL = (l == 0) ? 0 : (l == 1 ? (size_t)nN * 8 : (size_t)nN * 32);
#pragma unroll
      for (int r = 0; r < 8; ++r) {
        int ln = r + 8 * hi;
        int node = tileBase + ln;
        if (node >= nN) node = nN - 1;
        const float* Lx = L + ln * 144 + LOFF[l] + o * nm;
        float* po = out + baseL + (size_t)node * (CCH * nm) + o * nm;
#pragma unroll
        for (int m = 0; m < 2 * l + 1; ++m) po[m] = acc[m][r] + Lx[m];
      }
    }
  }
}

// ---------------- launch ----------------------------------------------------
extern "C" void kernel_launch(void* const* d_in, const int* in_sizes, int n_in,
                              void* d_out, int out_size, void* d_ws, size_t ws_size,
                              hipStream_t stream) {
  const float* x0 = (const float*)d_in[0];
  const float* x1 = (const float*)d_in[1];
  const float* x2 = (const float*)d_in[2];
  const float* ev = (const float*)d_in[3];
  const int* esrc = (const int*)d_in[4];
  const int* edst = (const int*)d_in[5];
  const float* wyy0 = (const float*)d_in[6];
  const float* wyy1 = (const float*)d_in[7];
  const float* wyy2 = (const float*)d_in[8];
  const float* wyx0 = (const float*)d_in[9];
  const float* wyx1 = (const float*)d_in[10];
  const float* wyx2 = (const float*)d_in[11];
  const float* wxx0 = (const float*)d_in[12];
  const float* wxx1 = (const float*)d_in[13];
  const float* wxx2 = (const float*)d_in[14];
  float* out = (float*)d_out;
  float* y   = (float*)d_ws;  // [nN][72] accumulator

  const int nN = in_sizes[0] / CCH;   // 20000
  const int nE = in_sizes[3];         // 320000

  // 1) zero accumulator (required every call: graph replays reuse d_ws)
  long ny = (long)nN * 72;
  zero_f32<<<(unsigned)((ny + 511) / 512), 512, 0, stream>>>(y, ny);

  // 2) message passing: 6 threads per edge, 3 float4 slots each
  long tot = (long)nE * 6;
  edge_mp<<<(unsigned)((tot + 255) / 256), 256, 0, stream>>>(x0, x1, x2, ev, esrc,
                                                             edst, y, nE);

  // 3) fused CG products + SO3 linear + residual, 32 nodes per block
  int blocks = (nN + 31) / 32;
  node_cg_wmma<<<blocks, 64, 0, stream>>>(x0, x1, x2, y, wyy0, wyy1, wyy2, wyx0,
                                          wyx1, wyx2, wxx0, wxx1, wxx2, out, nN);
}